// WFSA_21234318312108
// MI455X (gfx1250) — compile-verified
//
#include <hip/hip_runtime.h>
#include <hip/hip_bf16.h>

// Problem constants (match reference)
#define XS 512     // states
#define YS 128     // symbol feature dim
#define VS 128     // vocab
#define TS 512     // sequence length
#define NWG 16     // persistent scan workgroups

typedef __attribute__((ext_vector_type(2))) float v2f;
typedef __attribute__((ext_vector_type(8))) float v8f;
typedef __attribute__((ext_vector_type(4))) unsigned int u32x4;
typedef __attribute__((ext_vector_type(8))) int i32x8;
typedef __attribute__((ext_vector_type(4))) int i32x4;

#define M_ELEMS ((size_t)VS * XS * XS)            // 128 * 512 * 512
#define M_BYTES (M_ELEMS * sizeof(float))         // 128 MB
#define XBUF_OFF M_BYTES                          // 2 * 512 floats
#define CNT_OFF (M_BYTES + 4096)                  // barrier counter
#define AI_LDS_BYTES ((size_t)YS * XS * 4)        // 256 KB panel in LDS

// -----------------------------------------------------------------------------
// Kernel 1: M[v,i,j] = sum_y phi[v,y] * A[i,y,j]
// One block per i (512 blocks, 256 threads = 8 waves).
// The 128x512 A_i panel is DMA'd into LDS once via the Tensor Data Mover
// (tensor_load_to_lds, TENSORcnt), then all 8 waves build WMMA B fragments
// from LDS (removes the 8x redundant L2 reads of round 1).
// Wave w computes v-tile [16w, 16w+16) x all j via V_WMMA_F32_16X16X4_F32.
// -----------------------------------------------------------------------------
__global__ void __launch_bounds__(256)
wfsa_build_M(const float* __restrict__ A, const float* __restrict__ phi,
             float* __restrict__ Mout) {
  extern __shared__ float sA[];          // 128*512 floats (dynamic, 256 KB)

  const int i    = blockIdx.x;           // 0..511
  const int tid  = threadIdx.x;
  const int lane = tid & 31;             // wave32
  const int wave = tid >> 5;             // 0..7 -> v tile
  const int lr   = lane & 15;            // 0..15
  const int half = lane >> 4;            // 0 or 1
  const int v0   = wave * 16;

  // ---- Wave 0: issue TDM load of A_i (128 rows x 512 f32) into LDS ----
  if (tid < 32) {
    const unsigned lds_base = (unsigned)(uintptr_t)(&sA[0]);
    const unsigned long long ga =
        (unsigned long long)(uintptr_t)(A + (size_t)i * (YS * XS));
    u32x4 g0;
    g0.x = 1u;                                        // count=1 (valid), user mode
    g0.y = lds_base;                                  // lds_addr [63:32]
    g0.z = (unsigned)(ga & 0xFFFFFFFFu);              // global_addr lo
    g0.w = (unsigned)((ga >> 32) & 0x01FFFFFFu)       // global_addr [56:32]
         | (2u << 30);                                // type = 2 ("image")
    i32x8 g1;
    g1[0] = (int)(2u << 16);        // workgroup_mask=0, data_size=2 (4 bytes)
    g1[1] = (int)(512u << 16);      // tensor_dim0[15:0]=512  (row length j)
    g1[2] = (int)(128u << 16);      // tensor_dim0 hi=0 | tensor_dim1[15:0]=128
    g1[3] = (int)(512u << 16);      // tensor_dim1 hi=0 | tile_dim0=512
    g1[4] = 128;                    // tile_dim1=128, tile_dim2=0 (2D)
    g1[5] = 512;                    // tensor_dim0_stride lo (512 elements)
    g1[6] = 0;                      // stride hi / tensor_dim1_stride (unused)
    g1[7] = 0;
    const i32x4 z4 = {0, 0, 0, 0};
#if defined(__clang_major__) && __clang_major__ >= 23
    const i32x8 z8 = {0, 0, 0, 0, 0, 0, 0, 0};
    __builtin_amdgcn_tensor_load_to_lds(g0, g1, z4, z4, z8, 0);
#else
    __builtin_amdgcn_tensor_load_to_lds(g0, g1, z4, z4, 0);
#endif
  }

  // ---- Overlap: preload all 32 phi (A-operand) fragments from global ----
  // 16x4 f32 layout: lane l<16: VGPR0=phi[v0+l, y0], VGPR1=phi[v0+l, y0+1];
  // lanes 16-31 hold K offset +2.
  v2f pa[32];
  const float* prow = phi + (size_t)(v0 + lr) * YS;
#pragma unroll
  for (int k = 0; k < 32; ++k) {
    const int y = k * 4 + half * 2;
    pa[k].x = prow[y];
    pa[k].y = prow[y + 1];
  }

  // ---- Wait for the tensor DMA, publish the panel to all waves ----
  if (tid < 32) __builtin_amdgcn_s_wait_tensorcnt(0);
  __syncthreads();

  for (int jt = 0; jt < 32; ++jt) {
    const int j0 = jt * 16;
    v8f acc = {};                                  // zero C
#pragma unroll
    for (int k = 0; k < 32; ++k) {
      const int y = k * 4 + half * 2;
      // B-operand (A_i) fragment from LDS: 4x16 f32, rows K=y, cols N=lr
      v2f b;
      b.x = sA[(size_t)y * XS + j0 + lr];
      b.y = sA[(size_t)(y + 1) * XS + j0 + lr];
      acc = __builtin_amdgcn_wmma_f32_16x16x4_f32(
          /*neg_a=*/false, pa[k], /*neg_b=*/false, b,
          /*c_mod=*/(short)0, acc, /*reuse_a=*/false, /*reuse_b=*/false);
    }
    // C/D layout: VGPR r -> row r (lanes 0-15) / row r+8 (lanes 16-31), col = lr
#pragma unroll
    for (int r = 0; r < 8; ++r) {
      const int vrow = v0 + r + half * 8;
      Mout[((size_t)vrow * XS + i) * XS + (j0 + lr)] = acc[r];
    }
  }
}

// -----------------------------------------------------------------------------
// Kernel 2: persistent scan. 16 WGs x 256 threads. WG w owns columns
// [32w, 32w+32). Per step: x'[j] = sum_i x[i] * M[v_t][i,j]; global barrier.
// Prefetches next step's M rows (input sequence known ahead of time).
// -----------------------------------------------------------------------------
__global__ void __launch_bounds__(256)
wfsa_scan(const float* __restrict__ M, const int* __restrict__ inp,
          const float* __restrict__ xinit, const float* __restrict__ xfinal,
          float* __restrict__ xbuf, unsigned* __restrict__ cnt,
          float* __restrict__ out) {
  __shared__ float sx[XS];
  __shared__ float sred[256];

  const int tid  = threadIdx.x;
  const int wg   = blockIdx.x;
  const int lane = tid & 31;
  const int ww   = tid >> 5;           // 0..7 (i-chunk of 64)
  const int j    = wg * 32 + lane;     // owned column
  const int i0   = ww * 64;

  for (int t = 0; t < TS; ++t) {
    const int v = inp[t];
    const float* Mb = M + (size_t)v * (XS * XS);

    // Prefetch next step's rows for our column slice (independent of x).
    if (t + 1 < TS) {
      const int vn = inp[t + 1];
      const float* Mn = M + (size_t)vn * (XS * XS) + (size_t)wg * 32;
      __builtin_prefetch(Mn + (size_t)(i0 + lane) * XS, 0, 3);
      __builtin_prefetch(Mn + (size_t)(i0 + 32 + lane) * XS, 0, 3);
    }

    const float* xin = (t == 0) ? xinit : (xbuf + (size_t)((t & 1) ^ 1) * XS);
    sx[tid]       = xin[tid];
    sx[tid + 256] = xin[tid + 256];
    __syncthreads();

    float accv = 0.f;
    const float* Mcol = Mb + j;
#pragma unroll 8
    for (int k = 0; k < 64; ++k) {
      accv += sx[i0 + k] * Mcol[(size_t)(i0 + k) * XS];
    }
    sred[tid] = accv;
    __syncthreads();

    if (ww == 0) {                      // wave 0 reduces 8 partials per column
      float s = 0.f;
#pragma unroll
      for (int w2 = 0; w2 < 8; ++w2) s += sred[w2 * 32 + lane];
      xbuf[(size_t)(t & 1) * XS + j] = s;
    }

    // Device-wide sense barrier (release/acquire via L2 atomics).
    __threadfence();
    __syncthreads();
    if (tid == 0) {
      __threadfence();
      atomicAdd(cnt, 1u);
      const unsigned target = (unsigned)(t + 1) * NWG;
      while (__hip_atomic_load(cnt, __ATOMIC_RELAXED,
                               __HIP_MEMORY_SCOPE_AGENT) < target) { }
      __threadfence();
    }
    __syncthreads();
  }

  // Final: out = x_T . final  (x_T lives in buffer (TS-1)&1 == 1)
  if (wg == 0) {
    const float* xf = xbuf + XS;
    float p = xf[tid] * xfinal[tid] + xf[tid + 256] * xfinal[tid + 256];
    sred[tid] = p;
    __syncthreads();
    for (int s = 128; s > 0; s >>= 1) {
      if (tid < s) sred[tid] += sred[tid + s];
      __syncthreads();
    }
    if (tid == 0) out[0] = sred[0];
  }
}

// -----------------------------------------------------------------------------
extern "C" void kernel_launch(void* const* d_in, const int* in_sizes, int n_in,
                              void* d_out, int out_size, void* d_ws, size_t ws_size,
                              hipStream_t stream) {
  (void)in_sizes; (void)n_in; (void)out_size; (void)ws_size;

  const int*   inp    = (const int*)d_in[0];    // [512] int32
  const float* A      = (const float*)d_in[1];  // [512,128,512]
  const float* phi    = (const float*)d_in[2];  // [128,128]
  const float* xinit  = (const float*)d_in[3];  // [512]
  const float* xfinal = (const float*)d_in[4];  // [512]

  float*    M    = (float*)d_ws;                         // 128 MB
  float*    xbuf = (float*)((char*)d_ws + XBUF_OFF);     // 2*512 floats
  unsigned* cnt  = (unsigned*)((char*)d_ws + CNT_OFF);   // barrier counter
  float*    out  = (float*)d_out;

  // Reset barrier counter each call (graph-capture-safe async memset).
  hipMemsetAsync((char*)d_ws + CNT_OFF, 0, 128, stream);

  wfsa_build_M<<<XS, 256, (unsigned)AI_LDS_BYTES, stream>>>(A, phi, M);
  wfsa_scan<<<NWG, 256, 0, stream>>>(M, inp, xinit, xfinal, xbuf, cnt, out);
}